// ClusterNet_86079734546792
// MI455X (gfx1250) — compile-verified
//
#include <hip/hip_runtime.h>
#include <math.h>

// ---------------------------------------------------------------------------
// CDNA5 (gfx1250) implementation of the ClusterNet forward pass.
// f32 WMMA (V_WMMA_F32_16X16X4_F32) for all GEMM-shaped stages.
// ---------------------------------------------------------------------------

typedef float v2f __attribute__((ext_vector_type(2)));
typedef float v8f __attribute__((ext_vector_type(8)));

#define WMMA_F32X4(a, b, c) \
  __builtin_amdgcn_wmma_f32_16x16x4_f32(false, (a), false, (b), (short)0, (c), false, false)

// ---------------- conv1: [256,3,32,32] -> [256,6,28,28] --------------------
__global__ void conv1_kernel(const float* __restrict__ inp, const float* __restrict__ w,
                             const float* __restrict__ bias, float* __restrict__ out) {
  int i = blockIdx.x * blockDim.x + threadIdx.x;
  if (i >= 256 * 28 * 28) return;
  int ox = i % 28, oy = (i / 28) % 28, b = i / 784;
  const float* ib = inp + b * 3 * 1024;
  for (int oc = 0; oc < 6; ++oc) {
    float acc = bias[oc];
    for (int ic = 0; ic < 3; ++ic) {
      const float* ip = ib + ic * 1024 + oy * 32 + ox;
      const float* wp = w + (oc * 3 + ic) * 25;
#pragma unroll
      for (int ky = 0; ky < 5; ++ky)
#pragma unroll
        for (int kx = 0; kx < 5; ++kx)
          acc += ip[ky * 32 + kx] * wp[ky * 5 + kx];
    }
    out[(b * 6 + oc) * 784 + oy * 28 + ox] = acc;
  }
}

// ---------------- conv2: [256,6,14,14] -> [256,16,10,10] -------------------
__global__ void conv2_kernel(const float* __restrict__ x, const float* __restrict__ w,
                             const float* __restrict__ bias, float* __restrict__ out) {
  int i = blockIdx.x * blockDim.x + threadIdx.x;
  if (i >= 256 * 100) return;
  int ox = i % 10, oy = (i / 10) % 10, b = i / 100;
  const float* xb = x + b * 6 * 196;
  for (int oc = 0; oc < 16; ++oc) {
    float acc = bias[oc];
    for (int ic = 0; ic < 6; ++ic) {
      const float* ip = xb + ic * 196 + oy * 14 + ox;
      const float* wp = w + (oc * 6 + ic) * 25;
#pragma unroll
      for (int ky = 0; ky < 5; ++ky)
#pragma unroll
        for (int kx = 0; kx < 5; ++kx)
          acc += ip[ky * 14 + kx] * wp[ky * 5 + kx];
    }
    out[(b * 16 + oc) * 100 + oy * 10 + ox] = acc;
  }
}

// ------------- BatchNorm training stats: one block per channel -------------
__global__ void bn_stats_kernel(const float* __restrict__ x, int C, int HW, int Bn,
                                float* __restrict__ mean, float* __restrict__ rstd) {
  int c = blockIdx.x;
  __shared__ float ss[256], sq[256];
  float s = 0.f, q = 0.f;
  for (int b = 0; b < Bn; ++b) {
    const float* p = x + (b * C + c) * HW;
    for (int r = threadIdx.x; r < HW; r += 256) {
      float v = p[r];
      s += v; q += v * v;
    }
  }
  ss[threadIdx.x] = s; sq[threadIdx.x] = q;
  __syncthreads();
  for (int st = 128; st > 0; st >>= 1) {
    if (threadIdx.x < st) {
      ss[threadIdx.x] += ss[threadIdx.x + st];
      sq[threadIdx.x] += sq[threadIdx.x + st];
    }
    __syncthreads();
  }
  if (threadIdx.x == 0) {
    float P = (float)(Bn * HW);
    float m = ss[0] / P;
    float v = sq[0] / P - m * m;
    mean[c] = m;
    rstd[c] = rsqrtf(v + 1e-5f);
  }
}

__global__ void bn_apply_kernel(float* __restrict__ x, const float* __restrict__ mean,
                                const float* __restrict__ rstd, const float* __restrict__ g,
                                const float* __restrict__ bb, int C, int HW, int N) {
  int i = blockIdx.x * blockDim.x + threadIdx.x;
  if (i >= N) return;
  int c = (i / HW) % C;
  x[i] = (x[i] - mean[c]) * rstd[c] * g[c] + bb[c];
}

// ------- codebook logits 1 (K=6, pure streaming; store-BW bound) -----------
__global__ void logits1_kernel(const float* __restrict__ act, const float* __restrict__ keys,
                               float* __restrict__ out) {
  int i = blockIdx.x * blockDim.x + threadIdx.x;
  if (i >= 256 * 64 * 784) return;
  int r = i % 784;
  int k = (i / 784) % 64;
  int b = i / (784 * 64);
  const float* a = act + b * 6 * 784 + r;
  const float* kp = keys + k * 6;
  float acc = 0.f;
#pragma unroll
  for (int c = 0; c < 6; ++c) {
    float d = a[c * 784] - kp[c];
    acc += d * d;
  }
  out[i] = -sqrtf(acc);
}

// ---------------- relu + 2x2 maxpool ---------------------------------------
__global__ void pool_relu_kernel(const float* __restrict__ x, float* __restrict__ out,
                                 int H, int W, int N) {
  int i = blockIdx.x * blockDim.x + threadIdx.x;
  if (i >= N) return;
  int OW = W >> 1, OH = H >> 1;
  int ox = i % OW, oy = (i / OW) % OH, bc = i / (OW * OH);
  const float* p = x + bc * H * W + (oy * 2) * W + ox * 2;
  float m = fmaxf(fmaxf(p[0], p[1]), fmaxf(p[W], p[W + 1]));
  out[i] = fmaxf(m, 0.f);
}

// ---------------- norm helpers ---------------------------------------------
__global__ void rownorm_kernel(const float* __restrict__ x, float* __restrict__ out,
                               int R, int C) {
  int r = blockIdx.x * blockDim.x + threadIdx.x;
  if (r >= R) return;
  const float* p = x + r * C;
  float s = 0.f;
  for (int c = 0; c < C; ++c) s += p[c] * p[c];
  out[r] = s;
}

// act2 is [B,16,10,10]; position m = b*100 + r, channel stride 100.
__global__ void anorm2_kernel(const float* __restrict__ act2, float* __restrict__ out) {
  int m = blockIdx.x * blockDim.x + threadIdx.x;
  if (m >= 25600) return;
  int b = m / 100, r = m - b * 100;
  const float* p = act2 + b * 1600 + r;
  float s = 0.f;
#pragma unroll
  for (int c = 0; c < 16; ++c) { float v = p[c * 100]; s += v * v; }
  out[m] = s;
}

// ------- codebook logits 2 via WMMA: M=25600, N=64, K=16 -------------------
// dist^2 = |a|^2 - 2 a.k + |k|^2 ; the a.k GEMM runs on v_wmma_f32_16x16x4_f32
__global__ void logits2_wmma_kernel(const float* __restrict__ act2,
                                    const float* __restrict__ k2s,
                                    const float* __restrict__ anorm,
                                    const float* __restrict__ knorm,
                                    float* __restrict__ out) {
  int wid = (blockIdx.x * blockDim.x + threadIdx.x) >> 5;  // 0..6399
  int lane = threadIdx.x & 31;
  int tn = wid & 3;        // 4 N tiles (64 keys)
  int tm = wid >> 2;       // 1600 M tiles
  int col = lane & 15;
  int half = lane >> 4;

  int mA = tm * 16 + col;                // A-fragment row for this lane
  int bA = mA / 100, rA = mA - bA * 100;
  const float* ap = act2 + bA * 1600 + rA;       // A(m,k) = ap[k*100]
  int n = tn * 16 + col;
  const float* kp = k2s + n * 16;                // B(k,n) = kp[k]

  v8f c = {0.f, 0.f, 0.f, 0.f, 0.f, 0.f, 0.f, 0.f};
#pragma unroll
  for (int k0 = 0; k0 < 16; k0 += 4) {
    int ka = k0 + 2 * half;
    v2f a, b;
    a.x = ap[ka * 100];
    a.y = ap[(ka + 1) * 100];
    b.x = kp[ka];
    b.y = kp[ka + 1];
    c = WMMA_F32X4(a, b, c);
  }
  float kn = knorm[n];
#pragma unroll
  for (int j = 0; j < 8; ++j) {
    int m = tm * 16 + j + 8 * half;
    int b2 = m / 100, r2 = m - b2 * 100;
    float d2 = anorm[m] - 2.f * c[j] + kn;
    out[b2 * 6400 + n * 100 + r2] = -sqrtf(fmaxf(d2, 0.f));
  }
}

// ------- fc1 via WMMA: [256,400] x [400,120] + bias ------------------------
__global__ void fc1_wmma_kernel(const float* __restrict__ A, const float* __restrict__ W,
                                const float* __restrict__ bias, float* __restrict__ out) {
  int wid = (blockIdx.x * blockDim.x + threadIdx.x) >> 5;  // 0..127
  int lane = threadIdx.x & 31;
  int tn = wid & 7;        // N padded 120->128 => 8 tiles
  int tm = wid >> 3;       // 16 M tiles
  int col = lane & 15;
  int half = lane >> 4;

  const float* ap = A + (tm * 16 + col) * 400;
  int n = tn * 16 + col;
  bool nval = n < 120;
  const float* wp = W + (nval ? n : 0) * 400;

  v8f c = {0.f, 0.f, 0.f, 0.f, 0.f, 0.f, 0.f, 0.f};
  for (int k0 = 0; k0 < 400; k0 += 4) {
    int ka = k0 + 2 * half;
    v2f a, b;
    a.x = ap[ka];
    a.y = ap[ka + 1];
    b.x = nval ? wp[ka] : 0.f;
    b.y = nval ? wp[ka + 1] : 0.f;
    c = WMMA_F32X4(a, b, c);
  }
  if (nval) {
    float bv = bias[n];
#pragma unroll
    for (int j = 0; j < 8; ++j) {
      int m = tm * 16 + j + 8 * half;
      out[m * 120 + n] = c[j] + bv;
    }
  }
}

// ------- codebook logits 3 via WMMA: M=256, N=512, K=120 -------------------
__global__ void logits3_wmma_kernel(const float* __restrict__ act3,
                                    const float* __restrict__ k3s,
                                    const float* __restrict__ anorm,
                                    const float* __restrict__ knorm,
                                    float* __restrict__ out) {
  int wid = (blockIdx.x * blockDim.x + threadIdx.x) >> 5;  // 0..511
  int lane = threadIdx.x & 31;
  int tn = wid & 31;       // 32 N tiles (512 keys)
  int tm = wid >> 5;       // 16 M tiles
  int col = lane & 15;
  int half = lane >> 4;

  const float* ap = act3 + (tm * 16 + col) * 120;
  int n = tn * 16 + col;
  const float* kp = k3s + n * 120;

  v8f c = {0.f, 0.f, 0.f, 0.f, 0.f, 0.f, 0.f, 0.f};
  for (int k0 = 0; k0 < 120; k0 += 4) {
    int ka = k0 + 2 * half;
    v2f a, b;
    a.x = ap[ka];
    a.y = ap[ka + 1];
    b.x = kp[ka];
    b.y = kp[ka + 1];
    c = WMMA_F32X4(a, b, c);
  }
  float kn = knorm[n];
#pragma unroll
  for (int j = 0; j < 8; ++j) {
    int m = tm * 16 + j + 8 * half;
    float d2 = anorm[m] - 2.f * c[j] + kn;
    out[m * 512 + n] = -sqrtf(fmaxf(d2, 0.f));
  }
}

// ------- fc2 via WMMA: relu(act3)[256,120] x [120,10] + bias ---------------
__global__ void fc2_wmma_kernel(const float* __restrict__ act3, const float* __restrict__ W,
                                const float* __restrict__ bias, float* __restrict__ out) {
  int wid = (blockIdx.x * blockDim.x + threadIdx.x) >> 5;  // 0..15 (M tiles)
  int lane = threadIdx.x & 31;
  int tm = wid;
  int col = lane & 15;
  int half = lane >> 4;

  const float* ap = act3 + (tm * 16 + col) * 120;
  int n = col;                 // single N tile, pad 10 -> 16
  bool nval = n < 10;
  const float* wp = W + (nval ? n : 0) * 120;

  v8f c = {0.f, 0.f, 0.f, 0.f, 0.f, 0.f, 0.f, 0.f};
  for (int k0 = 0; k0 < 120; k0 += 4) {
    int ka = k0 + 2 * half;
    v2f a, b;
    a.x = fmaxf(ap[ka], 0.f);        // ReLU fused into A-fragment load
    a.y = fmaxf(ap[ka + 1], 0.f);
    b.x = nval ? wp[ka] : 0.f;
    b.y = nval ? wp[ka + 1] : 0.f;
    c = WMMA_F32X4(a, b, c);
  }
  if (nval) {
    float bv = bias[n];
#pragma unroll
    for (int j = 0; j < 8; ++j) {
      int m = tm * 16 + j + 8 * half;
      out[m * 10 + n] = c[j] + bv;
    }
  }
}

// ------- neural gas: per-row bitonic sort + argmax (1 block / batch row) ---
__global__ void neural_gas_kernel(const float* __restrict__ logits3,
                                  float* __restrict__ assign_out,
                                  float* __restrict__ rowloss) {
  __shared__ float sv[512];
  __shared__ float rv[256];
  __shared__ int ri[256];
  int row = blockIdx.x, t = threadIdx.x;
  const float* src = logits3 + row * 512;
  float v0 = src[t], v1 = src[t + 256];
  sv[t] = v0; sv[t + 256] = v1;

  // argmax with first-index tie-break (matches jnp.argmax)
  float mv; int mi;
  if (v0 >= v1) { mv = v0; mi = t; } else { mv = v1; mi = t + 256; }
  rv[t] = mv; ri[t] = mi;
  __syncthreads();
  for (int st = 128; st > 0; st >>= 1) {
    if (t < st) {
      float a = rv[t], b = rv[t + st];
      if (b > a || (b == a && ri[t + st] < ri[t])) { rv[t] = b; ri[t] = ri[t + st]; }
    }
    __syncthreads();
  }
  if (t == 0) assign_out[row] = (float)ri[0];

  // bitonic sort ascending, 512 elements, 256 threads
  for (int k = 2; k <= 512; k <<= 1) {
    for (int j = k >> 1; j > 0; j >>= 1) {
      __syncthreads();
      for (int i = t; i < 512; i += 256) {
        int ixj = i ^ j;
        if (ixj > i) {
          bool up = (i & k) == 0;
          float a = sv[i], b = sv[ixj];
          if ((a > b) == up) { sv[i] = b; sv[ixj] = a; }
        }
      }
    }
  }
  __syncthreads();

  float acc = sv[t] * sv[t] * expf(-(float)t) +
              sv[t + 256] * sv[t + 256] * expf(-(float)(t + 256));
  rv[t] = acc;
  __syncthreads();
  for (int st = 128; st > 0; st >>= 1) {
    if (t < st) rv[t] += rv[t + st];
    __syncthreads();
  }
  if (t == 0) rowloss[row] = rv[0];
}

__global__ void loss_mean_kernel(const float* __restrict__ rowloss, float* __restrict__ out) {
  __shared__ float s[256];
  s[threadIdx.x] = rowloss[threadIdx.x];
  __syncthreads();
  for (int st = 128; st > 0; st >>= 1) {
    if (threadIdx.x < st) s[threadIdx.x] += s[threadIdx.x + st];
    __syncthreads();
  }
  if (threadIdx.x == 0) out[0] = s[0] * (1.f / 256.f);
}

// ---------------------------------------------------------------------------
extern "C" void kernel_launch(void* const* d_in, const int* in_sizes, int n_in,
                              void* d_out, int out_size, void* d_ws, size_t ws_size,
                              hipStream_t stream) {
  (void)in_sizes; (void)n_in; (void)out_size; (void)ws_size;
  const float* inp     = (const float*)d_in[0];
  const float* conv1_w = (const float*)d_in[1];
  const float* conv1_b = (const float*)d_in[2];
  const float* bn1_g   = (const float*)d_in[3];
  const float* bn1_b   = (const float*)d_in[4];
  const float* conv2_w = (const float*)d_in[5];
  const float* conv2_b = (const float*)d_in[6];
  const float* bn2_g   = (const float*)d_in[7];
  const float* bn2_b   = (const float*)d_in[8];
  const float* fc1_w   = (const float*)d_in[9];
  const float* fc1_b   = (const float*)d_in[10];
  const float* fc2_w   = (const float*)d_in[11];
  const float* fc2_b   = (const float*)d_in[12];
  const float* k1s     = (const float*)d_in[13];
  const float* k2s     = (const float*)d_in[14];
  const float* k3s     = (const float*)d_in[15];

  float* out = (float*)d_out;
  // output tuple layout (flat, in return order)
  float* o_assign = out;                       // [256]
  float* o_loss   = out + 256;                 // [1]
  float* o_act4   = out + 257;                 // [256,10]
  float* o_l1     = out + 2817;                // [256,64,28,28]
  float* o_l2     = out + 2817 + 12845056;     // [256,64,10,10]
  float* o_l3     = o_l2 + 1638400;            // [256,512]

  // workspace layout (floats)
  float* ws     = (float*)d_ws;
  float* act1   = ws;                  // 1,204,224   [256,6,28,28]
  float* p1     = act1 + 1204224;      //   301,056   [256,6,14,14]
  float* act2   = p1 + 301056;         //   409,600   [256,16,10,10]
  float* p2     = act2 + 409600;       //   102,400   [256,400]
  float* act3   = p2 + 102400;         //    30,720   [256,120]
  float* anorm2 = act3 + 30720;        //    25,600
  float* knorm2 = anorm2 + 25600;      //        64
  float* anorm3 = knorm2 + 64;         //       256
  float* knorm3 = anorm3 + 256;        //       512
  float* mean1  = knorm3 + 512;        //         6
  float* rstd1  = mean1 + 6;           //         6
  float* mean2  = rstd1 + 6;           //        16
  float* rstd2  = mean2 + 16;          //        16
  float* rowls  = rstd2 + 16;          //       256

  const int T = 256;

  // conv1 + bias
  conv1_kernel<<<(200704 + T - 1) / T, T, 0, stream>>>(inp, conv1_w, conv1_b, act1);
  // bn1 (training stats, biased var) applied in place
  bn_stats_kernel<<<6, T, 0, stream>>>(act1, 6, 784, 256, mean1, rstd1);
  bn_apply_kernel<<<(1204224 + T - 1) / T, T, 0, stream>>>(act1, mean1, rstd1, bn1_g, bn1_b,
                                                           6, 784, 1204224);
  // codebook logits 1 (streaming, store-BW bound)
  logits1_kernel<<<(12845056 + T - 1) / T, T, 0, stream>>>(act1, k1s, o_l1);
  // relu + maxpool -> [256,6,14,14]
  pool_relu_kernel<<<(301056 + T - 1) / T, T, 0, stream>>>(act1, p1, 28, 28, 301056);

  // conv2 + bias, bn2
  conv2_kernel<<<(25600 + T - 1) / T, T, 0, stream>>>(p1, conv2_w, conv2_b, act2);
  bn_stats_kernel<<<16, T, 0, stream>>>(act2, 16, 100, 256, mean2, rstd2);
  bn_apply_kernel<<<(409600 + T - 1) / T, T, 0, stream>>>(act2, mean2, rstd2, bn2_g, bn2_b,
                                                          16, 100, 409600);
  // codebook logits 2 via WMMA (|a|^2 - 2 a.k + |k|^2)
  anorm2_kernel<<<(25600 + T - 1) / T, T, 0, stream>>>(act2, anorm2);
  rownorm_kernel<<<1, T, 0, stream>>>(k2s, knorm2, 64, 16);
  logits2_wmma_kernel<<<800, T, 0, stream>>>(act2, k2s, anorm2, knorm2, o_l2);
  // relu + maxpool -> flatten [256,400]
  pool_relu_kernel<<<(102400 + T - 1) / T, T, 0, stream>>>(act2, p2, 10, 10, 102400);

  // fc1 via WMMA -> act3 [256,120]
  fc1_wmma_kernel<<<16, T, 0, stream>>>(p2, fc1_w, fc1_b, act3);
  // codebook logits 3 via WMMA
  rownorm_kernel<<<1, T, 0, stream>>>(act3, anorm3, 256, 120);
  rownorm_kernel<<<2, T, 0, stream>>>(k3s, knorm3, 512, 120);
  logits3_wmma_kernel<<<64, T, 0, stream>>>(act3, k3s, anorm3, knorm3, o_l3);
  // fc2 via WMMA (ReLU fused on A load) -> act4 [256,10]
  fc2_wmma_kernel<<<2, T, 0, stream>>>(act3, fc2_w, fc2_b, o_act4);

  // neural-gas loss + argmax assignments
  neural_gas_kernel<<<256, T, 0, stream>>>(o_l3, o_assign, rowls);
  loss_mean_kernel<<<1, T, 0, stream>>>(rowls, o_loss);
}